// RoIPooling2D_953482740202
// MI455X (gfx1250) — compile-verified
//
#include <hip/hip_runtime.h>
#include <cfloat>
#include <cstdint>

// RoIPooling2D on MI455X (gfx1250).
// Memory-bound op (~20 MB traffic, zero FLOPs): use the CDNA5 Tensor Data
// Mover to DMA each ROI crop (a strided 3D tile: CH_TILE channel planes of
// ch x cw rows) from global memory into LDS with ONE tensor_load_to_lds
// instruction per block, then adaptive-max-pool 7x7 bins out of LDS.

#define RP_C 256
#define RP_H 50
#define RP_W 50
#define RP_OUT 7
#define RP_SCALE 0.0625f
#define CH_TILE 4   // 4 * 50 * 50 * 4B = 40 KB LDS worst case

// Guarantee a wave-uniform value is physically in an SGPR (LLVM will not
// auto-readfirstlane VALU-resident values for inline-asm "s" constraints).
__device__ __forceinline__ uint32_t rfl(uint32_t v) {
    return (uint32_t)__builtin_amdgcn_readfirstlane((int)v);
}

// Issue one TDM 3D-tile load. Descriptor groups are built in fixed SGPRs
// (s[52:55] g0, s[56:63] g1, s[64:67] g2, s[68:71] g3) because clang's "s"
// constraint does not form 256-bit SGPR tuples for inline asm operands.
__device__ __forceinline__ void tdm_load_3d_tile(uint64_t ga,       // global byte addr of tile origin
                                                 uint32_t lds_addr, // LDS byte addr
                                                 uint32_t cw, uint32_t ch,
                                                 uint32_t nplanes,
                                                 uint32_t row_stride,    // elems
                                                 uint32_t plane_stride)  // elems
{
#if defined(__gfx1250__)
    const uint32_t g0_0 = 1u;                                   // count=1
    const uint32_t g0_1 = lds_addr;                             // lds_addr
    const uint32_t g0_2 = (uint32_t)ga;                         // global lo
    const uint32_t g0_3 = (uint32_t)((ga >> 32) & 0x01FFFFFFu)  // global hi
                        | 0x80000000u;                          // type=2
    const uint32_t g1_0 = 0x00020000u;                          // data_size=4B
    const uint32_t g1_1 = (cw & 0xFFFFu) << 16;                 // td0[15:0]
    const uint32_t g1_2 = (cw >> 16) | ((ch & 0xFFFFu) << 16);  // td0 hi|td1 lo
    const uint32_t g1_3 = (ch >> 16) | (cw << 16);              // td1 hi|tile0
    const uint32_t g1_4 = ch | (nplanes << 16);                 // tile1|tile2
    const uint32_t g1_5 = row_stride;                           // td0_stride lo
    const uint32_t g1_6 = (plane_stride & 0xFFFFu) << 16;       // td1_stride lo
    const uint32_t g1_7 = plane_stride >> 16;                   // td1_stride hi
    const uint32_t g2_0 = nplanes;                              // tensor_dim2

    asm volatile(
        "s_mov_b32 s52, %0\n\t"
        "s_mov_b32 s53, %1\n\t"
        "s_mov_b32 s54, %2\n\t"
        "s_mov_b32 s55, %3\n\t"
        "s_mov_b32 s56, %4\n\t"
        "s_mov_b32 s57, %5\n\t"
        "s_mov_b32 s58, %6\n\t"
        "s_mov_b32 s59, %7\n\t"
        "s_mov_b32 s60, %8\n\t"
        "s_mov_b32 s61, %9\n\t"
        "s_mov_b32 s62, %10\n\t"
        "s_mov_b32 s63, %11\n\t"
        "s_mov_b32 s64, %12\n\t"
        "s_mov_b32 s65, 0\n\t"
        "s_mov_b32 s66, 0\n\t"
        "s_mov_b32 s67, 0\n\t"
        "s_mov_b32 s68, 0\n\t"
        "s_mov_b32 s69, 0\n\t"
        "s_mov_b32 s70, 0\n\t"
        "s_mov_b32 s71, 0\n\t"
        "tensor_load_to_lds s[52:55], s[56:63], s[64:67], s[68:71]"
        :
        : "s"(rfl(g0_0)), "s"(rfl(g0_1)), "s"(rfl(g0_2)), "s"(rfl(g0_3)),
          "s"(rfl(g1_0)), "s"(rfl(g1_1)), "s"(rfl(g1_2)), "s"(rfl(g1_3)),
          "s"(rfl(g1_4)), "s"(rfl(g1_5)), "s"(rfl(g1_6)), "s"(rfl(g1_7)),
          "s"(rfl(g2_0))
        : "memory",
          "s52","s53","s54","s55","s56","s57","s58","s59",
          "s60","s61","s62","s63","s64","s65","s66","s67",
          "s68","s69","s70","s71");
#else
    (void)ga; (void)lds_addr; (void)cw; (void)ch; (void)nplanes;
    (void)row_stride; (void)plane_stride;
#endif
}

__device__ __forceinline__ void wait_tensorcnt0() {
#if defined(__gfx1250__)
#if __has_builtin(__builtin_amdgcn_s_wait_tensorcnt)
    __builtin_amdgcn_s_wait_tensorcnt(0);
#else
    asm volatile("s_wait_tensorcnt 0x0" ::: "memory");
#endif
#endif
}

__global__ __launch_bounds__(256)
void roipool2d_tdm_kernel(const float* __restrict__ x,
                          const float* __restrict__ rois,
                          float* __restrict__ out,
                          int n_roi) {
    __shared__ float tile[CH_TILE * RP_H * RP_W];

    const int roi = blockIdx.y;
    const int c0  = blockIdx.x * CH_TILE;
    const int tid = threadIdx.x;
    if (roi >= n_roi) return;

    // ---- ROI geometry (matches reference exactly; all wave-uniform) ----
    const float* r = rois + (size_t)roi * 5;
    const int b = (int)r[0];
    int x1 = (int)floorf(r[1] * RP_SCALE);
    int y1 = (int)floorf(r[2] * RP_SCALE);
    int x2 = (int)floorf(r[3] * RP_SCALE);
    int y2 = (int)floorf(r[4] * RP_SCALE);
    x1 = min(max(x1, 0), RP_W - 1);
    y1 = min(max(y1, 0), RP_H - 1);
    x2 = max(min(max(x2, 0), RP_W - 1), x1);
    y2 = max(min(max(y2, 0), RP_H - 1), y1);
    const int cw = x2 - x1 + 1;   // inclusive crop width  (1..50)
    const int ch = y2 - y1 + 1;   // inclusive crop height (1..50)
    const int plane = ch * cw;

    // ---- One TDM op moves the whole [CH_TILE][ch][cw] strided tile ----
    // (TDM ignores EXEC and issues once per wave -> only wave 0 issues it.)
    if (tid < 32) {
        const float* gsrc = x + (size_t)b * RP_C * RP_H * RP_W
                              + (size_t)c0 * (RP_H * RP_W)
                              + (size_t)y1 * RP_W + x1;
        tdm_load_3d_tile((uint64_t)(uintptr_t)gsrc,
                         (uint32_t)(size_t)(void*)&tile[0],
                         (uint32_t)cw, (uint32_t)ch, (uint32_t)CH_TILE,
                         (uint32_t)RP_W, (uint32_t)(RP_H * RP_W));
        wait_tensorcnt0();         // wave 0's TDM writes to LDS complete
    }
    __syncthreads();               // publish LDS tile to all 8 waves

    // ---- Adaptive max-pool 7x7 from LDS ----
    for (int o = tid; o < CH_TILE * RP_OUT * RP_OUT; o += blockDim.x) {
        const int cc = o / (RP_OUT * RP_OUT);
        const int ij = o - cc * (RP_OUT * RP_OUT);
        const int i  = ij / RP_OUT;
        const int j  = ij - i * RP_OUT;
        // bin [floor(i*h/7), ceil((i+1)*h/7)) -- always non-empty for h>=1
        const int ys = (i * ch) / RP_OUT;
        const int ye = ((i + 1) * ch + (RP_OUT - 1)) / RP_OUT;
        const int xs = (j * cw) / RP_OUT;
        const int xe = ((j + 1) * cw + (RP_OUT - 1)) / RP_OUT;

        float m = -FLT_MAX;
        const int base = cc * plane;
        for (int yy = ys; yy < ye; ++yy) {
            const int rowb = base + yy * cw;
            for (int xx = xs; xx < xe; ++xx)
                m = fmaxf(m, tile[rowb + xx]);
        }
        out[((((size_t)roi * RP_C) + (c0 + cc)) * RP_OUT + i) * RP_OUT + j] = m;
    }
}

extern "C" void kernel_launch(void* const* d_in, const int* in_sizes, int n_in,
                              void* d_out, int out_size, void* d_ws, size_t ws_size,
                              hipStream_t stream) {
    (void)n_in; (void)out_size; (void)d_ws; (void)ws_size;
    const float* x    = (const float*)d_in[0];
    const float* rois = (const float*)d_in[1];
    float* out        = (float*)d_out;
    const int n_roi   = in_sizes[1] / 5;                 // 128

    dim3 grid(RP_C / CH_TILE, (unsigned)n_roi);          // 64 x 128 blocks
    dim3 block(256);                                     // 8 wave32 waves
    hipLaunchKernelGGL(roipool2d_tdm_kernel, grid, block, 0, stream,
                       x, rois, out, n_roi);
}